// GNN_74990128988715
// MI455X (gfx1250) — compile-verified
//
#include <hip/hip_runtime.h>

#define N_NODES  50000
#define N_EDGES  600000
#define N_GRAPHS 512
#define DIM_IN   64
#define DIM_H    128

typedef float v2f __attribute__((ext_vector_type(2)));
typedef float v8f __attribute__((ext_vector_type(8)));

// ---------------- utility ----------------
__global__ void zero_f32(float* __restrict__ p, int n) {
    int i = blockIdx.x * blockDim.x + threadIdx.x;
    if (i < n) p[i] = 0.0f;
}

__global__ void deg_accum(const int* __restrict__ dst, float* __restrict__ deg) {
    int e = blockIdx.x * blockDim.x + threadIdx.x;
    if (e < N_EDGES) atomicAdd(&deg[dst[e]], 1.0f);
}

__global__ void deg_to_dinv(float* __restrict__ d) {
    int i = blockIdx.x * blockDim.x + threadIdx.x;
    if (i < N_NODES) d[i] = rsqrtf(d[i] + 1.0f);   // deg includes self-loop (+1)
}

// ---------------- dense GEMM on the matrix pipe ----------------
// out[16*gridDim.x x Hout] = A[.. x K] @ W[K x Hout], fp32 WMMA 16x16x4.
// One wave per 16x16 tile; blockDim.x == 32 so EXEC is all ones (WMMA req).
__global__ void gemm_wmma_f32(const float* __restrict__ A,
                              const float* __restrict__ W,
                              float* __restrict__ out,
                              int K, int Hout) {
    const int lane = threadIdx.x;          // 0..31
    const int half = lane >> 4;            // 0: lanes 0-15, 1: lanes 16-31
    const int l15  = lane & 15;
    const int koff = half * 2;             // A/B K sub-offset per ISA layout
    const int rowA = blockIdx.x * 16 + l15;
    const int colB = blockIdx.y * 16 + l15;
    const float* __restrict__ Ar = A + (size_t)rowA * K;

    v8f c = {};
    for (int k = 0; k < K; k += 4) {
        v2f a, b;
        a.x = Ar[k + koff];
        a.y = Ar[k + koff + 1];
        b.x = W[(size_t)(k + koff)     * Hout + colB];
        b.y = W[(size_t)(k + koff + 1) * Hout + colB];
        // 8 args: (neg_a, A, neg_b, B, c_mod, C, reuse_a, reuse_b)
        c = __builtin_amdgcn_wmma_f32_16x16x4_f32(false, a, false, b,
                                                  (short)0, c, false, false);
    }
    // D layout: VGPR r -> M=r (lanes 0-15) / M=r+8 (lanes 16-31); N = lane&15
    const int rowD = blockIdx.x * 16 + half * 8;
#pragma unroll
    for (int r = 0; r < 8; ++r)
        out[(size_t)(rowD + r) * Hout + colB] = c[r];
}

// ---------------- GCN aggregation ----------------
// agg = h * dinv^2 (self-loop term) -- also serves as the zero-init of agg
__global__ void self_loop_init(const float* __restrict__ h,
                               const float* __restrict__ dinv,
                               float* __restrict__ agg) {
    int i = blockIdx.x * blockDim.x + threadIdx.x;
    if (i >= N_NODES * DIM_H) return;
    int n = i >> 7;                        // DIM_H == 128
    float dv = dinv[n];
    agg[i] = h[i] * dv * dv;
}

// one wave per edge: 32 lanes x float4 covers all 128 features
__global__ void edge_scatter(const int* __restrict__ src,
                             const int* __restrict__ dst,
                             const float* __restrict__ dinv,
                             const float* __restrict__ h,
                             float* __restrict__ agg) {
    int tid = blockIdx.x * blockDim.x + threadIdx.x;
    int e   = tid >> 5;
    if (e >= N_EDGES) return;
    int grp = tid & 31;
    int s = src[e], d = dst[e];
    float norm = dinv[s] * dinv[d];
    const float4 hv =
        *reinterpret_cast<const float4*>(h + (size_t)s * DIM_H + grp * 4);
    float* ap = agg + (size_t)d * DIM_H + grp * 4;
    atomicAdd(ap + 0, hv.x * norm);
    atomicAdd(ap + 1, hv.y * norm);
    atomicAdd(ap + 2, hv.z * norm);
    atomicAdd(ap + 3, hv.w * norm);
}

__global__ void relu_bias(const float* __restrict__ agg,
                          const float* __restrict__ b,
                          float* __restrict__ h) {
    int i = blockIdx.x * blockDim.x + threadIdx.x;
    if (i >= N_NODES * DIM_H) return;
    h[i] = fmaxf(agg[i] + b[i & (DIM_H - 1)], 0.0f);
}

// ---------------- pooling + MLP head ----------------
__global__ void pool_accum(const int* __restrict__ batch,
                           const float* __restrict__ h,
                           float* __restrict__ psum,
                           float* __restrict__ cnt) {
    int i = blockIdx.x * blockDim.x + threadIdx.x;
    if (i >= N_NODES * DIM_H) return;
    int n = i >> 7, j = i & 127;
    int g = batch[n];
    atomicAdd(&psum[(size_t)g * DIM_H + j], h[i]);
    if (j == 0) atomicAdd(&cnt[g], 1.0f);
}

__global__ void inv_counts(float* __restrict__ cnt) {
    int g = blockIdx.x * blockDim.x + threadIdx.x;
    if (g < N_GRAPHS) cnt[g] = 1.0f / fmaxf(cnt[g], 1.0f);
}

__global__ void head1(const float* __restrict__ psum,
                      const float* __restrict__ invcnt,
                      const float* __restrict__ mW1,
                      const float* __restrict__ mb1,
                      float* __restrict__ z) {
    int g = blockIdx.x;           // G blocks
    int j = threadIdx.x;          // 64 threads
    float ic  = invcnt[g];
    float acc = mb1[j];
    const float* pr = psum + (size_t)g * DIM_H;
#pragma unroll 8
    for (int k = 0; k < DIM_H; ++k)
        acc += pr[k] * ic * mW1[(size_t)k * (DIM_H / 2) + j];
    z[(size_t)g * (DIM_H / 2) + j] = fmaxf(acc, 0.0f);
}

__global__ void head2(const float* __restrict__ z,
                      const float* __restrict__ mW2,
                      const float* __restrict__ mb2,
                      float* __restrict__ out) {
    int g = blockIdx.x * blockDim.x + threadIdx.x;
    if (g >= N_GRAPHS) return;
    float acc = mb2[0];
    const float* zr = z + (size_t)g * (DIM_H / 2);
#pragma unroll 8
    for (int j = 0; j < DIM_H / 2; ++j) acc += zr[j] * mW2[j];
    out[g] = acc;
}

// ---------------- launcher ----------------
extern "C" void kernel_launch(void* const* d_in, const int* in_sizes, int n_in,
                              void* d_out, int out_size, void* d_ws, size_t ws_size,
                              hipStream_t stream) {
    (void)in_sizes; (void)n_in; (void)out_size; (void)ws_size;

    const float* x     = (const float*)d_in[0];
    const int*   ei    = (const int*)d_in[1];
    const int*   src   = ei;
    const int*   dst   = ei + N_EDGES;
    const int*   batch = (const int*)d_in[2];
    const float* W0 = (const float*)d_in[3];  const float* b0 = (const float*)d_in[4];
    const float* W1 = (const float*)d_in[5];  const float* b1 = (const float*)d_in[6];
    const float* W2 = (const float*)d_in[7];  const float* b2 = (const float*)d_in[8];
    const float* mW1 = (const float*)d_in[9]; const float* mb1 = (const float*)d_in[10];
    const float* mW2 = (const float*)d_in[11];const float* mb2 = (const float*)d_in[12];
    float* out = (float*)d_out;

    char* wsb = (char*)d_ws;
    size_t off = 0;
    auto alloc = [&](size_t bytes) -> float* {
        float* p = (float*)(wsb + off);
        off = (off + bytes + 255) & ~(size_t)255;
        return p;
    };
    const size_t NH = (size_t)N_NODES * DIM_H;
    float* dinv = alloc((size_t)N_NODES * 4);
    float* bufA = alloc(NH * 4);                         // GEMM output (pre-agg)
    float* bufB = alloc(NH * 4);                         // aggregation accumulator
    float* bufC = alloc(NH * 4);                         // activated features
    float* psum = alloc((size_t)N_GRAPHS * DIM_H * 4);
    float* cnt  = alloc((size_t)N_GRAPHS * 4);
    float* z    = alloc((size_t)N_GRAPHS * (DIM_H / 2) * 4);

    const int nhThreads = N_NODES * DIM_H;               // 6.4M, fits int
    const int scThreads = N_EDGES * 32;                  // 19.2M, fits int
    dim3 gemmGrid(N_NODES / 16, DIM_H / 16);

    // degree -> dinv
    zero_f32<<<(N_NODES + 255) / 256, 256, 0, stream>>>(dinv, N_NODES);
    deg_accum<<<(N_EDGES + 255) / 256, 256, 0, stream>>>(dst, dinv);
    deg_to_dinv<<<(N_NODES + 255) / 256, 256, 0, stream>>>(dinv);

    const float* layer_in = x;
    const float* Ws[3] = {W0, W1, W2};
    const float* bs[3] = {b0, b1, b2};
    int          Ks[3] = {DIM_IN, DIM_H, DIM_H};
    for (int l = 0; l < 3; ++l) {
        gemm_wmma_f32<<<gemmGrid, 32, 0, stream>>>(layer_in, Ws[l], bufA, Ks[l], DIM_H);
        self_loop_init<<<(nhThreads + 255) / 256, 256, 0, stream>>>(bufA, dinv, bufB);
        edge_scatter<<<(scThreads + 255) / 256, 256, 0, stream>>>(src, dst, dinv, bufA, bufB);
        relu_bias<<<(nhThreads + 255) / 256, 256, 0, stream>>>(bufB, bs[l], bufC);
        layer_in = bufC;
    }

    // global mean pool
    zero_f32<<<(N_GRAPHS * DIM_H + 255) / 256, 256, 0, stream>>>(psum, N_GRAPHS * DIM_H);
    zero_f32<<<(N_GRAPHS + 255) / 256, 256, 0, stream>>>(cnt, N_GRAPHS);
    pool_accum<<<(nhThreads + 255) / 256, 256, 0, stream>>>(batch, bufC, psum, cnt);
    inv_counts<<<(N_GRAPHS + 255) / 256, 256, 0, stream>>>(cnt);

    // MLP head
    head1<<<N_GRAPHS, DIM_H / 2, 0, stream>>>(psum, cnt, mW1, mb1, z);
    head2<<<(N_GRAPHS + 255) / 256, 256, 0, stream>>>(z, mW2, mb2, out);
}